// baseGNN_31044023616097
// MI455X (gfx1250) — compile-verified
//
#include <hip/hip_runtime.h>

typedef float v2f __attribute__((ext_vector_type(2)));
typedef float v8f __attribute__((ext_vector_type(8)));

#define DIM 128

// ---------------------------------------------------------------------------
// GEMM: Out[n_rows,128] = (RELU ? relu(A) : A)[n_rows,128] @ W[128,128]
// One wave computes a 16-row x 128-col strip with V_WMMA_F32_16X16X4_F32.
// W is staged in LDS packed as (K,K+1) float2 pairs so each lane's B operand
// (VGPR0=K+2*half, VGPR1=K+2*half+1 per ISA 16x4/4x16 layout) is one b64 load.
// Requires n_rows % 16 == 0 (true here: 100000 = 6250*16).
// ---------------------------------------------------------------------------
template <int RELU>
__global__ __launch_bounds__(256) void gcn_gemm_wmma(
    const float* __restrict__ A, const float* __restrict__ W,
    float* __restrict__ Out, int n_rows)
{
    __shared__ v2f Wp[64 * DIM];   // 64 KB: pair-row kk holds (W[2kk][n], W[2kk+1][n])

    for (int i = threadIdx.x; i < 64 * DIM; i += 256) {
        int kk = i >> 7;
        int n  = i & (DIM - 1);
        v2f w;
        w.x = W[(2 * kk) * DIM + n];
        w.y = W[(2 * kk + 1) * DIM + n];
        Wp[i] = w;
    }
    __syncthreads();

    const int lane = threadIdx.x & 31;
    const int wave = threadIdx.x >> 5;
    const int row0 = (blockIdx.x * 8 + wave) * 16;
    if (row0 >= n_rows) return;            // wave-uniform: EXEC stays all-1s for WMMA

    const int mrow = row0 + (lane & 15);   // A row for this lane
    const int half = lane >> 4;            // K-half select per ISA layout
    const int ncol = lane & 15;            // C/D column within tile

    v8f acc[8] = {};                       // 8 N-tiles of 16 cols = 128 cols

    const float* arow = A + (size_t)mrow * DIM + 2 * half;
    #pragma unroll 4
    for (int k0 = 0; k0 < DIM; k0 += 4) {
        v2f a = *(const v2f*)(arow + k0);  // A[m][k0+2h], A[m][k0+2h+1]
        if (RELU) {
            a.x = a.x > 0.0f ? a.x : 0.0f;
            a.y = a.y > 0.0f ? a.y : 0.0f;
        }
        const v2f* brow = &Wp[((k0 >> 1) + half) * DIM + ncol];
        #pragma unroll
        for (int t = 0; t < 8; ++t) {
            // 8 args: (neg_a, A, neg_b, B, c_mod, C, reuse_a, reuse_b)
            acc[t] = __builtin_amdgcn_wmma_f32_16x16x4_f32(
                false, a, false, brow[t * 16], (short)0, acc[t], false, false);
        }
    }

    // C/D layout: VGPR r -> row (r + 8*half), col = lane&15 within each tile
    #pragma unroll
    for (int r = 0; r < 8; ++r) {
        float* orow = Out + (size_t)(row0 + r + 8 * half) * DIM + ncol;
        #pragma unroll
        for (int t = 0; t < 8; ++t) orow[t * 16] = acc[t][r];
    }
}

// ---------------------------------------------------------------------------
// Out[i][:] = b[:]  (bias pre-initialization of the scatter accumulator)
// ---------------------------------------------------------------------------
__global__ __launch_bounds__(256) void init_bias(
    float* __restrict__ Out, const float* __restrict__ b, int n_rows)
{
    int i = blockIdx.x * 256 + threadIdx.x;
    if (i < n_rows * DIM) Out[i] = b[i & (DIM - 1)];
}

// ---------------------------------------------------------------------------
// For each edge e: Out[dst[e]][:] += M[src[e]][:]
// One wave per edge: 32 lanes x float4 = 128 features. e is wave-uniform;
// readfirstlane lets the compiler turn the index fetches into scalar loads
// and the gather/atomic bases into SGPR (saddr) addressing. M and Out are
// both 51.2 MB -> L2-resident; the 80M atomics/layer resolve at L2.
// ---------------------------------------------------------------------------
__global__ __launch_bounds__(256) void scatter_add(
    const float* __restrict__ M, const int* __restrict__ src,
    const int* __restrict__ dst, float* __restrict__ Out, int n_edges)
{
    int e    = (blockIdx.x * 256 + threadIdx.x) >> 5;   // wave-uniform
    int lane = threadIdx.x & 31;
    e = __builtin_amdgcn_readfirstlane(e);              // force scalar
    if (e >= n_edges) return;

    const int s = __builtin_amdgcn_readfirstlane(src[e]);
    const int d = __builtin_amdgcn_readfirstlane(dst[e]);

    const float4 v = *(const float4*)(M + (size_t)s * DIM + lane * 4);
    float* o = Out + (size_t)d * DIM + lane * 4;
    unsafeAtomicAdd(o + 0, v.x);
    unsafeAtomicAdd(o + 1, v.y);
    unsafeAtomicAdd(o + 2, v.z);
    unsafeAtomicAdd(o + 3, v.w);
}

// ---------------------------------------------------------------------------
// Inputs: x[100000,128] f32, edge_index[2,625000] i32, W1, b1, W2, b2
// d_ws: m buffer [N,128] f32 (51.2 MB). d_out doubles as the h1 accumulator.
// ---------------------------------------------------------------------------
extern "C" void kernel_launch(void* const* d_in, const int* in_sizes, int n_in,
                              void* d_out, int out_size, void* d_ws, size_t ws_size,
                              hipStream_t stream) {
    const float* x  = (const float*)d_in[0];
    const int*   ei = (const int*)d_in[1];
    const float* W1 = (const float*)d_in[2];
    const float* b1 = (const float*)d_in[3];
    const float* W2 = (const float*)d_in[4];
    const float* b2 = (const float*)d_in[5];
    float* out = (float*)d_out;
    float* m   = (float*)d_ws;

    const int N = in_sizes[0] / DIM;   // 100000
    const int E = in_sizes[1] / 2;     // 625000
    const int* src = ei;
    const int* dst = ei + E;

    dim3 blk(256);
    const int gemm_blocks = (N / 16 + 7) / 8;
    const int init_blocks = (N * DIM + 255) / 256;
    const int scat_blocks = (E * 32 + 255) / 256;

    // Layer 1: m = x @ W1 ; out = b1 ; out += scatter(m)
    gcn_gemm_wmma<0><<<gemm_blocks, blk, 0, stream>>>(x, W1, m, N);
    init_bias      <<<init_blocks, blk, 0, stream>>>(out, b1, N);
    scatter_add    <<<scat_blocks, blk, 0, stream>>>(m, src, dst, out, E);

    // Layer 2: m = relu(out) @ W2 ; out = b2 ; out += scatter(m)
    gcn_gemm_wmma<1><<<gemm_blocks, blk, 0, stream>>>(out, W2, m, N);
    init_bias      <<<init_blocks, blk, 0, stream>>>(out, b2, N);
    scatter_add    <<<scat_blocks, blk, 0, stream>>>(m, src, dst, out, E);
}